// S2_GaussianBlur_15453292331293
// MI455X (gfx1250) — compile-verified
//
#include <hip/hip_runtime.h>
#include <cmath>

// S2_GaussianBlur: depthwise circular separable Gaussian blur.
// x: (1,6,2048,2048) fp32 -> out: (1,6,2048,2048) fp32.
// Memory-bound: 200MB @ 23.3TB/s ~ 8.6us floor. Horizontal pass on VALU into
// LDS, vertical pass as banded matmul on the matrix pipe via
// V_WMMA_F32_16X16X4_F32 (full-precision fp32 WMMA, K accumulated 4 at a time).

#define NRR 2048
#define NCC 2048
#define KSZ 10
#define NCH 6

#define TILE_R 16
#define NWAVES 8
#define TILE_C (16 * NWAVES)   // 128 columns per workgroup
#define TMP_ROWS 28            // 25 needed (16 + 10 - 1), padded to 7 chunks of K=4
#define LDS_PITCH 132          // break bank alignment (132*4 B rows)

typedef __attribute__((ext_vector_type(2))) float v2f;
typedef __attribute__((ext_vector_type(8))) float v8f;

struct TapsArg { float g[NCH][KSZ]; };

__global__ __launch_bounds__(256)
void s2_gaussian_blur_fused(const float* __restrict__ x,
                            float* __restrict__ out,
                            TapsArg taps) {
  __shared__ float s_tmp[TMP_ROWS * LDS_PITCH];
  __shared__ float s_g[16];   // taps[0..9], [10..15] = 0 (zero pad for band matrix)

  const int tid  = threadIdx.x;
  const int lane = tid & 31;
  const int wave = tid >> 5;
  const int ch   = blockIdx.z;
  const int r0   = blockIdx.y * TILE_R;
  const int c0   = blockIdx.x * TILE_C;

  const float* __restrict__ xc = x   + (size_t)ch * NRR * NCC;
  float* __restrict__       oc = out + (size_t)ch * NRR * NCC;

  if (tid < 16) s_g[tid] = (tid < KSZ) ? taps.g[ch][tid] : 0.0f;
  __syncthreads();

  // ---- Phase 1: horizontal pass into LDS ----------------------------------
  // tmp[row][col] for row = 0..24 mapping to image row (r0 - 4 + row) mod NR,
  // col = 0..127 mapping to image col c0+col. Rows 25..27 zeroed (K padding).
  // Consecutive tids -> consecutive cols -> coalesced loads per tap.
#pragma unroll
  for (int it = 0; it < (TMP_ROWS * TILE_C) / 256; ++it) {   // 3584/256 = 14
    int idx = tid + it * 256;
    int row = idx >> 7;        // / 128
    int col = idx & 127;
    float acc = 0.0f;
    if (row < 25) {
      int gr = (r0 - 4 + row) & (NRR - 1);                  // circular wrap
      const float* __restrict__ rp = xc + (size_t)gr * NCC;
#pragma unroll
      for (int u = 0; u < KSZ; ++u) {
        int gc = (c0 + col - 4 + u) & (NCC - 1);            // circular wrap
        acc = fmaf(s_g[u], rp[gc], acc);
      }
    }
    s_tmp[row * LDS_PITCH + col] = acc;
  }
  __syncthreads();

  // ---- Phase 2: vertical pass as banded matmul on WMMA --------------------
  // O(16x16) = A(16x28) * T(28x16), A[m,k] = g[k-m] for 0<=k-m<=9 else 0.
  // fp32 WMMA 16x16x4, 7 K-chunks. Per documented 16x4 fp32 A layout:
  //   lanes 0-15: M=lane, VGPR0=K(base+0), VGPR1=K(base+1); lanes 16-31: K(base+2/3)
  // B(4x16) assumed mirrored: VGPRv holds K = 2*(lane>>4) + v, N = lane&15.
  const int m  = lane & 15;     // M (for A) / N (for B) index
  const int lh = lane >> 4;     // which half of the wave
  const int cw = wave * 16;     // this wave's column slice in the tile

  v8f acc = {};
#pragma unroll
  for (int kk = 0; kk < 7; ++kk) {
    const int kb = 4 * kk + 2 * lh;
    // A fragment: band weights; indices outside [0,10) select zeroed s_g slots
    int d0 = kb - m;
    int d1 = kb + 1 - m;
    int i0 = ((unsigned)d0 < KSZ) ? d0 : 15;
    int i1 = ((unsigned)d1 < KSZ) ? d1 : 15;
    v2f a;
    a.x = s_g[i0];
    a.y = s_g[i1];
    // B fragment: intermediate rows kb, kb+1 (max index 27 < TMP_ROWS)
    v2f b;
    b.x = s_tmp[(kb + 0) * LDS_PITCH + cw + m];
    b.y = s_tmp[(kb + 1) * LDS_PITCH + cw + m];
    acc = __builtin_amdgcn_wmma_f32_16x16x4_f32(
        /*neg_a=*/false, a, /*neg_b=*/false, b,
        /*c_mod=*/(short)0, acc, /*reuse_a=*/false, /*reuse_b=*/false);
  }

  // ---- Store: C/D layout VGPRv -> (M=v, lanes0-15) / (M=v+8, lanes16-31) --
  const int cc = c0 + cw + m;
#pragma unroll
  for (int v = 0; v < 8; ++v) {
    int rr = r0 + v + 8 * lh;
    oc[(size_t)rr * NCC + cc] = acc[v];
  }
}

extern "C" void kernel_launch(void* const* d_in, const int* in_sizes, int n_in,
                              void* d_out, int out_size, void* d_ws, size_t ws_size,
                              hipStream_t stream) {
  (void)in_sizes; (void)n_in; (void)out_size; (void)d_ws; (void)ws_size;

  // Normalized separable taps; symmetric coords make the reference's flip a no-op:
  //   sigma = scale * sqrt(-2 ln(mtf) / pi^2),  ghat[t] = e_t / sum(e),
  //   e_t = exp(-(t-4.5)^2 / (2 sigma^2)).   (1/(sigma sqrt(2pi)) cancels.)
  static const double mtf[NCH] = {0.38, 0.34, 0.34, 0.26, 0.22, 0.23};
  TapsArg taps;
  for (int c = 0; c < NCH; ++c) {
    double sigma = 2.0 * std::sqrt(-2.0 * std::log(mtf[c])) / M_PI;
    double e[KSZ], s = 0.0;
    for (int t = 0; t < KSZ; ++t) {
      double d = (double)t - 4.5;
      e[t] = std::exp(-d * d / (2.0 * sigma * sigma));
      s += e[t];
    }
    for (int t = 0; t < KSZ; ++t) taps.g[c][t] = (float)(e[t] / s);
  }

  const float* x  = (const float*)d_in[0];
  float*       yo = (float*)d_out;

  dim3 grid(NCC / TILE_C, NRR / TILE_R, NCH);   // (16, 128, 6)
  s2_gaussian_blur_fused<<<grid, 256, 0, stream>>>(x, yo, taps);
}